// GCN_13718125543735
// MI455X (gfx1250) — compile-verified
//
#include <hip/hip_runtime.h>

#define NNODES 100000
#define NEDGES 1000000
#define D 64
// GROUPS == 3

typedef __attribute__((ext_vector_type(2))) float v2f;
typedef __attribute__((ext_vector_type(8))) float v8f;

// ---------------------------------------------------------------------------
// Zero the workspace accumulators (msg_sum [N*64] + deg [N]).
// ---------------------------------------------------------------------------
__global__ void gcn_zero_ws(float* __restrict__ ws, int n) {
  int i = blockIdx.x * blockDim.x + threadIdx.x;
  if (i < n) ws[i] = 0.0f;
}

// ---------------------------------------------------------------------------
// Edge scatter: one thread per (edge, dim). Feature table (25.6MB) and the
// accumulator both fit in the 192MB L2, whose atomic units absorb the RMWs.
// ---------------------------------------------------------------------------
__global__ void gcn_scatter(const float* __restrict__ feat,
                            const int* __restrict__ src,
                            const int* __restrict__ dst,
                            float* __restrict__ msg,
                            float* __restrict__ deg) {
  long long i = (long long)blockIdx.x * blockDim.x + threadIdx.x;
  if (i >= (long long)NEDGES * D) return;
  int e = (int)(i >> 6);
  int d = (int)(i & 63);
  int s = src[e];
  int t = dst[e];
  atomicAdd(&msg[(long long)t * D + d], feat[(long long)s * D + d]);
  if (d == 0) atomicAdd(&deg[t], 1.0f);
}

// ---------------------------------------------------------------------------
// Mean + group gating, one wave32 per node. Lane l owns dims l and l+32.
// ge_g = relu(h . W_gc[:,g] + b_gc[g]); h *= #(g : ge_g == max_g ge_g).
// h is written in-place over msg_sum.
// ---------------------------------------------------------------------------
__global__ void gcn_gate(float* __restrict__ msg,
                         const float* __restrict__ deg,
                         const float* __restrict__ Wgc,   // [64,3] row-major
                         const float* __restrict__ bgc) { // [3]
  int lane = threadIdx.x & 31;
  int wid  = threadIdx.x >> 5;
  int node = blockIdx.x * (blockDim.x >> 5) + wid;
  if (node >= NNODES) return;

  float dg  = deg[node];
  float inv = dg > 0.0f ? 1.0f / dg : 0.0f;
  long long base = (long long)node * D;
  float h0 = msg[base + lane]      * inv;
  float h1 = msg[base + lane + 32] * inv;

  float g0 = h0 * Wgc[lane * 3 + 0] + h1 * Wgc[(lane + 32) * 3 + 0];
  float g1 = h0 * Wgc[lane * 3 + 1] + h1 * Wgc[(lane + 32) * 3 + 1];
  float g2 = h0 * Wgc[lane * 3 + 2] + h1 * Wgc[(lane + 32) * 3 + 2];

#pragma unroll
  for (int off = 16; off > 0; off >>= 1) {
    g0 += __shfl_xor(g0, off, 32);
    g1 += __shfl_xor(g1, off, 32);
    g2 += __shfl_xor(g2, off, 32);
  }
  g0 = fmaxf(g0 + bgc[0], 0.0f);
  g1 = fmaxf(g1 + bgc[1], 0.0f);
  g2 = fmaxf(g2 + bgc[2], 0.0f);
  float top = fmaxf(fmaxf(g0, g1), g2);
  float cnt = (g0 == top ? 1.0f : 0.0f) +
              (g1 == top ? 1.0f : 0.0f) +
              (g2 == top ? 1.0f : 0.0f);

  msg[base + lane]      = h0 * cnt;
  msg[base + lane + 32] = h1 * cnt;
}

// ---------------------------------------------------------------------------
// out[N,64] = h[N,64] @ W_lin[64,64]^T + b_lin, via V_WMMA_F32_16X16X4_F32.
// One wave per 16x16 output tile; block = 32 threads so EXEC is all ones.
// A layout (16x4 f32): lane = M + 16*(K/2), vgpr = K%2.
// B layout (4x16 f32): lane = N + 16*(K/2), vgpr = K%2; B[k,o] = W_lin[o,k].
// C/D (16x16 f32):     lane = N + 16*(M/8), vgpr = M%8.
// ---------------------------------------------------------------------------
__global__ void gcn_wmma_linear(const float* __restrict__ h,
                                const float* __restrict__ Wl,  // [64,64]
                                const float* __restrict__ bl,  // [64]
                                float* __restrict__ out) {
  int lane = threadIdx.x & 31;
  int tile = blockIdx.x;     // (N/16)*4 tiles
  int mt = tile >> 2;
  int ot = tile & 3;
  int m0 = mt * 16;
  int o0 = ot * 16;
  int half = lane >> 4;      // selects K-pair (A) / K-pair (B) / M-half (C)
  int l16  = lane & 15;

  const float* hrow = h  + (long long)(m0 + l16) * D;  // A row (M = l16)
  const float* wrow = Wl + (long long)(o0 + l16) * D;  // B col (N = l16)

  v8f c = {};
#pragma unroll
  for (int kb = 0; kb < D; kb += 4) {
    int k = kb + half * 2;
    v2f a, b;
    a.x = hrow[k];
    a.y = hrow[k + 1];
    b.x = wrow[k];
    b.y = wrow[k + 1];
    c = __builtin_amdgcn_wmma_f32_16x16x4_f32(
        /*neg_a=*/false, a, /*neg_b=*/false, b,
        /*c_mod=*/(short)0, c, /*reuse_a=*/false, /*reuse_b=*/false);
  }

  float bias = bl[o0 + l16];
#pragma unroll
  for (int v = 0; v < 8; ++v) {
    int m = m0 + v + half * 8;
    out[(long long)m * D + (o0 + l16)] = c[v] + bias;
  }
}

// ---------------------------------------------------------------------------
extern "C" void kernel_launch(void* const* d_in, const int* in_sizes, int n_in,
                              void* d_out, int out_size, void* d_ws, size_t ws_size,
                              hipStream_t stream) {
  const float* feat = (const float*)d_in[0];  // [N, 64]
  const int*   src  = (const int*)d_in[1];    // [E]
  const int*   dst  = (const int*)d_in[2];    // [E]
  const float* Wgc  = (const float*)d_in[3];  // [64, 3]
  const float* bgc  = (const float*)d_in[4];  // [1, 3]
  const float* Wl   = (const float*)d_in[5];  // [64, 64]
  const float* bl   = (const float*)d_in[6];  // [64]
  float* out = (float*)d_out;                 // [N, 64]

  float* msg = (float*)d_ws;                  // [N*64]
  float* deg = msg + (long long)NNODES * D;   // [N]

  // 1) zero accumulators
  {
    int n = NNODES * (D + 1);
    gcn_zero_ws<<<(n + 255) / 256, 256, 0, stream>>>(msg, n);
  }
  // 2) edge scatter-add
  {
    long long total = (long long)NEDGES * D;
    int blocks = (int)((total + 255) / 256);
    gcn_scatter<<<blocks, 256, 0, stream>>>(feat, src, dst, msg, deg);
  }
  // 3) mean + gating (8 waves per 256-thread block, one wave per node)
  {
    int blocks = (NNODES + 7) / 8;
    gcn_gate<<<blocks, 256, 0, stream>>>(msg, deg, Wgc, bgc);
  }
  // 4) WMMA linear layer: one wave per 16x16 tile
  {
    int blocks = (NNODES / 16) * 4;  // 100000 % 16 == 0
    gcn_wmma_linear<<<blocks, 32, 0, stream>>>(msg, Wl, bl, out);
  }
}